// separate_encoder_84911503442683
// MI455X (gfx1250) — compile-verified
//
#include <hip/hip_runtime.h>
#include <hip/hip_bf16.h>
#include <math.h>

typedef __attribute__((ext_vector_type(2))) float v2f;
typedef __attribute__((ext_vector_type(8))) float v8f;
typedef __attribute__((address_space(3))) float lds_float;

// ---------------------------------------------------------------------------
// prep: per-chunk mean over the step axis; div = x - mean;
// mean_in = raw x for chunk 0, broadcast mean for chunks > 0.
// x viewed as [B*cnt, step, 512]; one thread per (b,chunk,d) column.
// ---------------------------------------------------------------------------
__global__ __launch_bounds__(256) void prep_chunks(
    const float* __restrict__ x, float* __restrict__ mean_in,
    float* __restrict__ divv, int total /* B*cnt*512 */, int cnt, int step)
{
    int idx = blockIdx.x * 256 + threadIdx.x;
    if (idx >= total) return;
    int d  = idx & 511;
    int bc = idx >> 9;
    int chunk = bc % cnt;
    const float* xp = x + (size_t)bc * step * 512 + d;
    float s = 0.f;
    for (int l = 0; l < step; ++l) s += xp[(size_t)l * 512];
    float mean = s / (float)step;
    float* mp = mean_in + (size_t)bc * step * 512 + d;
    float* dp = divv    + (size_t)bc * step * 512 + d;
    for (int l = 0; l < step; ++l) {
        float xv = xp[(size_t)l * 512];
        dp[(size_t)l * 512] = xv - mean;
        mp[(size_t)l * 512] = (chunk == 0) ? xv : mean;
    }
}

// ---------------------------------------------------------------------------
// Y[M,512] = X[M,512] @ W^T + bias (+ resid), W row-major [512,512].
// Block: 256 threads = 8 wave32s, tile BM=128 x BN=64, BK=16.
// fp32 WMMA V_WMMA_F32_16X16X4_F32; tiles staged with double-buffered
// GLOBAL_LOAD_ASYNC_TO_LDS (ASYNCcnt) so DMA of slab k+16 overlaps WMMA on k.
//   Xs: [2][128][20] floats -> 80B row stride (16B aligned for b128 async,
//       stride-20 columns hit 16 distinct banks mod 64).
//   BsP: K-pairs as float2 so a B fragment is one aligned ds_load_b64.
// A layout: lanes 0-15 M=lane (K pair 0,1), lanes 16-31 M=lane-16 (K pair 2,3).
// B layout: VGPR0 = row k (lanes 0-15) / row k+2 (lanes 16-31), VGPR1 = +1.
// C layout: VGPR r = row r (lanes 0-15) / row r+8 (lanes 16-31), col = lane%16.
// M must be a multiple of 128 (147456 and 49152 both are).
// ---------------------------------------------------------------------------
__global__ __launch_bounds__(256) void gemm512_wmma(
    const float* __restrict__ X, const float* __restrict__ W,
    const float* __restrict__ bias, const float* __restrict__ resid,
    float* __restrict__ Y, int M)
{
    __shared__ float Xs[2][128][20];
    __shared__ v2f   BsP[2][8][65];
    const int tid  = threadIdx.x;
    const int lane = tid & 31;
    const int wave = tid >> 5;
    const int m0 = blockIdx.x * 128;
    const int n0 = blockIdx.y * 64;

    const v8f vzero = {0.f, 0.f, 0.f, 0.f, 0.f, 0.f, 0.f, 0.f};
    v8f acc[4];
#pragma unroll
    for (int nt = 0; nt < 4; ++nt) acc[nt] = vzero;

    // async DMA staging of one 16-wide K slab into LDS buffer `buf`
    auto stage = [&](int buf, int k0) {
#pragma unroll
        for (int i = 0; i < 2; ++i) {          // X: 128x16 floats, b128/lane
            int r = (tid >> 2) + i * 64;       // 0..127
            int c = (tid & 3) * 4;             // 0,4,8,12
            unsigned l_off = (unsigned)(size_t)(lds_float*)&Xs[buf][r][c];
            unsigned long long g =
                (unsigned long long)(size_t)&X[(size_t)(m0 + r) * 512 + k0 + c];
            asm volatile("global_load_async_to_lds_b128 %0, %1, off"
                         :: "v"(l_off), "v"(g) : "memory");
        }
#pragma unroll
        for (int i = 0; i < 2; ++i) {          // W: 64x16 floats as K-pairs, b64/lane
            int idx = tid + i * 256;
            int n  = idx & 63;
            int kq = idx >> 6;                 // 0..7
            unsigned l_off = (unsigned)(size_t)(lds_float*)&BsP[buf][kq][n];
            unsigned long long g =
                (unsigned long long)(size_t)&W[(size_t)(n0 + n) * 512 + k0 + kq * 2];
            asm volatile("global_load_async_to_lds_b64 %0, %1, off"
                         :: "v"(l_off), "v"(g) : "memory");
        }
    };

    stage(0, 0);
    asm volatile("s_wait_asynccnt 0" ::: "memory");
    __syncthreads();

    const int mrow = wave * 16 + (lane & 15);
    const int kblk = (lane >> 4) * 2;          // 0 for lanes 0-15, 2 for 16-31
    const int ncol = lane & 15;

    for (int k0 = 0; k0 < 512; k0 += 16) {
        const int buf = (k0 >> 4) & 1;
        if (k0 + 16 < 512) stage(buf ^ 1, k0 + 16);   // overlap DMA with WMMA
#pragma unroll
        for (int kk = 0; kk < 16; kk += 4) {
            v2f a = *(const v2f*)&Xs[buf][mrow][kk + kblk];
            const int kq = (kk + kblk) >> 1;          // per-lane K pair index
#pragma unroll
            for (int nt = 0; nt < 4; ++nt) {
                v2f b = BsP[buf][kq][nt * 16 + ncol];
                acc[nt] = __builtin_amdgcn_wmma_f32_16x16x4_f32(
                    false, a, false, b, (short)0, acc[nt], false, false);
            }
        }
        asm volatile("s_wait_asynccnt 0" ::: "memory");
        __syncthreads();
    }

    const int rbase = m0 + wave * 16 + ((lane >> 4) << 3);
#pragma unroll
    for (int nt = 0; nt < 4; ++nt) {
        int col = n0 + nt * 16 + ncol;
        float bv = bias[col];
#pragma unroll
        for (int r = 0; r < 8; ++r) {
            size_t off = (size_t)(rbase + r) * 512 + col;
            float val = acc[nt][r] + bv;
            if (resid) val += resid[off];
            Y[off] = val;
        }
    }
}

// ---------------------------------------------------------------------------
// Fused per-(sequence, head) attention: scores = QK^T/8, softmax, A@V.
// L <= 24, E = 64. AO may alias Q (each block reads its tile into LDS before
// writing back its own tile only), so Q/AO are NOT restrict-qualified.
// ---------------------------------------------------------------------------
__global__ __launch_bounds__(128) void attn_fused(
    const float* Q, const float* __restrict__ K,
    const float* __restrict__ V, float* AO, int L)
{
    __shared__ float Qs[24][64];
    __shared__ float Ks[24][64];
    __shared__ float Vs[24][64];
    __shared__ float Ss[24][25];
    const int n = blockIdx.x, h = blockIdx.y;
    const int tid = threadIdx.x;
    const size_t base = (size_t)n * L * 512 + (size_t)h * 64;

    for (int idx = tid; idx < L * 64; idx += 128) {
        int l = idx >> 6, e = idx & 63;
        size_t g = base + (size_t)l * 512 + e;
        Qs[l][e] = Q[g];
        Ks[l][e] = K[g];
        Vs[l][e] = V[g];
    }
    __syncthreads();

    const float scale = 0.125f;           // 1/sqrt(64)
    for (int idx = tid; idx < L * L; idx += 128) {
        int i = idx / L, j = idx % L;
        float s = 0.f;
#pragma unroll
        for (int e = 0; e < 64; ++e) s += Qs[i][e] * Ks[j][e];
        Ss[i][j] = s * scale;
    }
    __syncthreads();

    if (tid < L) {                        // row-wise softmax
        float mx = -1e30f;
        for (int j = 0; j < L; ++j) mx = fmaxf(mx, Ss[tid][j]);
        float sum = 0.f;
        for (int j = 0; j < L; ++j) {
            float e = __expf(Ss[tid][j] - mx);
            Ss[tid][j] = e; sum += e;
        }
        float inv = 1.f / sum;
        for (int j = 0; j < L; ++j) Ss[tid][j] *= inv;
    }
    __syncthreads();

    for (int idx = tid; idx < L * 64; idx += 128) {
        int l = idx >> 6, e = idx & 63;
        float o = 0.f;
        for (int j = 0; j < L; ++j) o += Ss[l][j] * Vs[j][e];
        AO[base + (size_t)l * 512 + e] = o;
    }
}

// ---------------------------------------------------------------------------
// out[n,m,d] = ELU( sum_l y[n,l,d] * Wl[m,l] + bl[m] ), one thread per (n,d).
// ---------------------------------------------------------------------------
template <int L, int LO>
__global__ __launch_bounds__(256) void seqlin_elu(
    const float* __restrict__ y, const float* __restrict__ Wl,
    const float* __restrict__ bl, float* __restrict__ out,
    int total /* Nseq*512 */)
{
    int idx = blockIdx.x * 256 + threadIdx.x;
    if (idx >= total) return;
    int d = idx & 511, n = idx >> 9;
    float yl[L];
#pragma unroll
    for (int l = 0; l < L; ++l) yl[l] = y[((size_t)n * L + l) * 512 + d];
#pragma unroll
    for (int m = 0; m < LO; ++m) {
        float s = bl[m];
#pragma unroll
        for (int l = 0; l < L; ++l) s += Wl[m * L + l] * yl[l];
        s = (s > 0.f) ? s : expm1f(s);    // ELU(alpha=1)
        out[((size_t)n * LO + m) * 512 + d] = s;
    }
}

// ---------------------------------------------------------------------------
// irfft(rfft(o)*conj(rfft(o))) == circular autocorrelation:
//   r[j] = sum_l o[l] * o[(l+j) mod Lo];  out = o + t*r for chunks > 0.
// ---------------------------------------------------------------------------
template <int LO>
__global__ __launch_bounds__(256) void autocorr_fuse(
    const float* __restrict__ in, const float* __restrict__ tscale,
    float* __restrict__ out, int total /* B*cnt*512 */, int cnt)
{
    int idx = blockIdx.x * 256 + threadIdx.x;
    if (idx >= total) return;
    int d = idx & 511, bc = idx >> 9;
    int chunk = bc % cnt;
    const float* ip = in  + (size_t)bc * LO * 512 + d;
    float*       op = out + (size_t)bc * LO * 512 + d;
    float v[LO];
#pragma unroll
    for (int j = 0; j < LO; ++j) v[j] = ip[(size_t)j * 512];
    if (chunk == 0) {
#pragma unroll
        for (int j = 0; j < LO; ++j) op[(size_t)j * 512] = v[j];
    } else {
        float t = tscale[0];
#pragma unroll
        for (int j = 0; j < LO; ++j) {
            float r = 0.f;
#pragma unroll
            for (int l = 0; l < LO; ++l) r += v[l] * v[(l + j) & (LO - 1)];
            op[(size_t)j * 512] = v[j] + t * r;
        }
    }
}

// ---------------------------------------------------------------------------
// Host orchestration
// ---------------------------------------------------------------------------
struct EncP {
    const float *Wk, *Wl, *Wo, *Wq, *Wv, *bk, *bl, *bo, *bq, *bv;
};

// jax pytree flatten order within one encoder dict (keys sorted):
// Wk, Wl, Wo, Wq, Wv, bk, bl, bo, bq, bv
static EncP getp(void* const* d_in, int i0) {
    EncP p;
    p.Wk = (const float*)d_in[i0 + 0];
    p.Wl = (const float*)d_in[i0 + 1];
    p.Wo = (const float*)d_in[i0 + 2];
    p.Wq = (const float*)d_in[i0 + 3];
    p.Wv = (const float*)d_in[i0 + 4];
    p.bk = (const float*)d_in[i0 + 5];
    p.bl = (const float*)d_in[i0 + 6];
    p.bo = (const float*)d_in[i0 + 7];
    p.bq = (const float*)d_in[i0 + 8];
    p.bv = (const float*)d_in[i0 + 9];
    return p;
}

static void run_branch(const float* Xin, const EncP& P, const float* tptr,
                       float* Qb, float* Kb, float* Vb, float* dest,
                       int Nseq, int L, int Lo, int cnt, hipStream_t stream)
{
    const int M = Nseq * L;               // multiple of 128
    dim3 gg(M / 128, 8);
    gemm512_wmma<<<gg, 256, 0, stream>>>(Xin, P.Wq, P.bq, nullptr, Qb, M);
    gemm512_wmma<<<gg, 256, 0, stream>>>(Xin, P.Wk, P.bk, nullptr, Kb, M);
    gemm512_wmma<<<gg, 256, 0, stream>>>(Xin, P.Wv, P.bv, nullptr, Vb, M);
    attn_fused<<<dim3(Nseq, 8), 128, 0, stream>>>(Qb, Kb, Vb, Qb, L);
    // y = Xin + AO @ Wo^T + bo   (AO lives in Qb, y goes to Kb)
    gemm512_wmma<<<gg, 256, 0, stream>>>(Qb, P.Wo, P.bo, Xin, Kb, M);
    const int tot = Nseq * 512;           // == B*cnt*512
    if (L == 24)
        seqlin_elu<24, 8><<<(tot + 255) / 256, 256, 0, stream>>>(Kb, P.Wl, P.bl, Vb, tot);
    else
        seqlin_elu<8, 4><<<(tot + 255) / 256, 256, 0, stream>>>(Kb, P.Wl, P.bl, Vb, tot);
    if (Lo == 8)
        autocorr_fuse<8><<<(tot + 255) / 256, 256, 0, stream>>>(Vb, tptr, dest, tot, cnt);
    else
        autocorr_fuse<4><<<(tot + 255) / 256, 256, 0, stream>>>(Vb, tptr, dest, tot, cnt);
}

extern "C" void kernel_launch(void* const* d_in, const int* in_sizes, int n_in,
                              void* d_out, int out_size, void* d_ws, size_t ws_size,
                              hipStream_t stream)
{
    (void)in_sizes; (void)n_in; (void)out_size; (void)ws_size;
    const float* x = (const float*)d_in[0];
    EncP p01 = getp(d_in, 1);    // layers[0].b1
    EncP p02 = getp(d_in, 11);   // layers[0].b2
    EncP p11 = getp(d_in, 21);   // layers[1].b1
    EncP p12 = getp(d_in, 31);   // layers[1].b2
    const float* t1 = (const float*)d_in[41];
    const float* t2 = (const float*)d_in[42];

    float* ws = (float*)d_ws;
    const size_t SZ = (size_t)147456 * 512;  // layer-0 activation size (floats)
    float* A  = ws;                          // mean_in
    float* Dv = ws + SZ;                     // div
    float* Qb = ws + 2 * SZ;                 // Q, then attention output
    float* Kb = ws + 3 * SZ;                 // K, then y (residual+O-proj)
    float* Vb = ws + 4 * SZ;                 // V, then seq-linear output
    float* X1 = ws + 5 * SZ;                 // layer-1 input [64,768,512]

    float* out    = (float*)d_out;           // tuple: (x, div0, div1)
    float* out_x  = out;                                         // [64,384,512]
    float* out_d0 = out + (size_t)64 * 384 * 512;                // [64,768,512]
    float* out_d1 = out + (size_t)64 * 384 * 512 + (size_t)64 * 768 * 512;

    // ---- layer 0: S=2304, step=24, cnt=96, L=24, Lo=8 ----
    {
        const int cnt = 96, step = 24, Nseq = 6144;
        const int tot = 64 * cnt * 512;
        prep_chunks<<<(tot + 255) / 256, 256, 0, stream>>>(x, A, Dv, tot, cnt, step);
        run_branch(A,  p01, t1, Qb, Kb, Vb, X1,     Nseq, 24, 8, cnt, stream);
        run_branch(Dv, p02, t2, Qb, Kb, Vb, out_d0, Nseq, 24, 8, cnt, stream);
    }
    // ---- layer 1: S=768, step=8, cnt=96, L=8, Lo=4 ----
    {
        const int cnt = 96, step = 8, Nseq = 6144;
        const int tot = 64 * cnt * 512;
        prep_chunks<<<(tot + 255) / 256, 256, 0, stream>>>(X1, A, Dv, tot, cnt, step);
        run_branch(A,  p11, t1, Qb, Kb, Vb, out_x,  Nseq, 8, 4, cnt, stream);
        run_branch(Dv, p12, t2, Qb, Kb, Vb, out_d1, Nseq, 8, 4, cnt, stream);
    }
}